// ChainCRF_60129542144448
// MI455X (gfx1250) — compile-verified
//
#include <hip/hip_runtime.h>
#include <hip/hip_bf16.h>

#define B_   256
#define S_   1024
#define T_   64
#define WSTR 68   // LDS row stride in floats: conflict-free for A-reads and D-stores

#define LN2_F     0.69314718056f
#define INV_LN2_F 1.44269504089f

typedef float v2f __attribute__((ext_vector_type(2)));
typedef float v8f __attribute__((ext_vector_type(8)));

// ---------------------------------------------------------------- setup ----
__global__ void crf_setup(const float* __restrict__ U, float* __restrict__ expU,
                          float* __restrict__ out) {
    int tid = threadIdx.x;
    if (tid == 0) out[0] = 0.0f;               // d_out is poisoned; must init
    for (int i = tid; i < T_ * T_; i += blockDim.x)
        expU[i] = __expf(U[i]);
}

// ----------------------------------------------------------- path energy ----
__global__ void crf_path(const float* __restrict__ emis, const int* __restrict__ tags,
                         const float* __restrict__ U, const float* __restrict__ bs,
                         const float* __restrict__ be, float* __restrict__ path_e) {
    const int b = blockIdx.x, tid = threadIdx.x;
    float acc = 0.0f;
    for (int t = tid; t < S_; t += blockDim.x) {
        int tg = tags[b * S_ + t];
        float v = emis[(size_t)(b * S_ + t) * T_ + tg];
        if (t == 0)      v += bs[tg];
        if (t == S_ - 1) v += be[tg];
        if (t + 1 < S_)  v += U[tg * T_ + tags[b * S_ + t + 1]];
        acc += v;
    }
    __shared__ float red[256];
    red[tid] = acc; __syncthreads();
    for (int s = 128; s > 0; s >>= 1) {
        if (tid < s) red[tid] += red[tid + s];
        __syncthreads();
    }
    if (tid == 0) path_e[b] = red[0];
}

// ------------------------------------------------------- forward recursion ----
// Grid: B/16 blocks. Block: 128 threads = 4 wave32; wave w owns state columns
// [16w, 16w+16). alpha kept as w = exp(alpha - O[row]) in LDS (16 x 64, stride 68).
__launch_bounds__(128)
__global__ void crf_forward(const float* __restrict__ emis,
                            const float* __restrict__ expU,
                            const float* __restrict__ bs,
                            const float* __restrict__ be,
                            const float* __restrict__ path_e,
                            float* __restrict__ out) {
    __shared__ float lds_w[16 * WSTR];
    __shared__ float wmax[4][16];
    __shared__ float Ofs[16];
    __shared__ float red[16];

    const int tid   = threadIdx.x;
    const int lane  = tid & 31;
    const int wv    = tid >> 5;        // wave id = N-tile id (0..3)
    const int half  = lane >> 4;       // lane half (0/1)
    const int l15   = lane & 15;
    const int jc    = wv * 16 + l15;   // state column for this lane (D layout)
    const int rbase = half * 8;        // row base for this lane (D layout)
    const int bblk  = blockIdx.x * 16;

    // B operands (expU columns for this wave): 16 x v2f, loop-invariant.
    // Layout mirrors A (16x4 f32): component0 = K {0|2}+4kt, component1 = K {1|3}+4kt.
    v2f breg[16];
    {
        const int kb = half * 2;
        #pragma unroll
        for (int kt = 0; kt < 16; ++kt) {
            int k0 = kt * 4 + kb;
            v2f bb; bb.x = expU[k0 * T_ + jc]; bb.y = expU[(k0 + 1) * T_ + jc];
            breg[kt] = bb;
        }
    }
    const float bsv = bs[jc];
    const float bev = be[jc];

    // ---- t = 0: d = x0 + b_start (D layout: 8 rows per lane) ----
    float d[8];
    #pragma unroll
    for (int r = 0; r < 8; ++r)
        d[r] = emis[((size_t)(bblk + rbase + r) * S_ + 0) * T_ + jc] + bsv;

    // Per-row max over this wave's 16 columns (xor masks 1..8 stay in the half)
    float m[8];
    #pragma unroll
    for (int r = 0; r < 8; ++r) {
        float v = d[r];
        v = fmaxf(v, __shfl_xor(v, 1));
        v = fmaxf(v, __shfl_xor(v, 2));
        v = fmaxf(v, __shfl_xor(v, 4));
        v = fmaxf(v, __shfl_xor(v, 8));
        m[r] = v;
    }
    if (l15 == 0) {
        #pragma unroll
        for (int r = 0; r < 8; ++r) wmax[wv][rbase + r] = m[r];
    }
    __syncthreads();
    float dm[8];
    #pragma unroll
    for (int r = 0; r < 8; ++r) {
        float v = fmaxf(fmaxf(wmax[0][rbase + r], wmax[1][rbase + r]),
                        fmaxf(wmax[2][rbase + r], wmax[3][rbase + r]));
        dm[r] = v;
    }
    if (tid < 16) {
        Ofs[tid] = fmaxf(fmaxf(wmax[0][tid], wmax[1][tid]),
                         fmaxf(wmax[2][tid], wmax[3][tid]));
    }
    #pragma unroll
    for (int r = 0; r < 8; ++r)
        lds_w[(rbase + r) * WSTR + jc] =
            __builtin_amdgcn_exp2f((d[r] - dm[r]) * INV_LN2_F);
    __syncthreads();

    // Prefetch x for t=1 (D layout registers)
    float xc[8];
    #pragma unroll
    for (int r = 0; r < 8; ++r)
        xc[r] = emis[((size_t)(bblk + rbase + r) * S_ + 1) * T_ + jc];

    // A-operand base: lane l<16 reads row l, K pair (4kt,4kt+1); lane>=16 row l-16, (4kt+2,4kt+3)
    const float* wrowA = &lds_w[l15 * WSTR + half * 2];

    for (int t = 1; t < S_; ++t) {
        // deep prefetch (t+8) into cache: lowers to global_prefetch_b8
        if (t + 8 < S_) {
            const float* p = &emis[((size_t)(bblk + rbase) * S_ + (t + 8)) * T_ + jc];
            #pragma unroll
            for (int r = 0; r < 8; ++r)
                __builtin_prefetch(p + (size_t)r * S_ * T_, 0, 1);
        }

        float xn[8];
        #pragma unroll
        for (int r = 0; r < 8; ++r) xn[r] = 0.0f;
        if (t + 1 < S_) {
            #pragma unroll
            for (int r = 0; r < 8; ++r)
                xn[r] = emis[((size_t)(bblk + rbase + r) * S_ + (t + 1)) * T_ + jc];
        }

        // s = w @ expU : chained f32 WMMA, K = 64 via 16 x 16x16x4
        v8f c = {0.f, 0.f, 0.f, 0.f, 0.f, 0.f, 0.f, 0.f};
        #pragma unroll
        for (int kt = 0; kt < 16; ++kt) {
            v2f a = *(const v2f*)(wrowA + kt * 4);
            c = __builtin_amdgcn_wmma_f32_16x16x4_f32(
                    false, a, false, breg[kt], (short)0, c, false, false);
        }

        // d = log(s) + x (+ b_end on last step). s in [0.8, 80] -> raw v_log_f32 ok.
        #pragma unroll
        for (int r = 0; r < 8; ++r) {
            float x = xc[r];
            if (t == S_ - 1) x += bev;
            d[r] = fmaf(__builtin_amdgcn_logf(c[r]), LN2_F, x);
        }

        #pragma unroll
        for (int r = 0; r < 8; ++r) {
            float v = d[r];
            v = fmaxf(v, __shfl_xor(v, 1));
            v = fmaxf(v, __shfl_xor(v, 2));
            v = fmaxf(v, __shfl_xor(v, 4));
            v = fmaxf(v, __shfl_xor(v, 8));
            m[r] = v;
        }
        if (l15 == 0) {
            #pragma unroll
            for (int r = 0; r < 8; ++r) wmax[wv][rbase + r] = m[r];
        }
        __syncthreads();   // wmax visible; also fences all lds_w reads above

        #pragma unroll
        for (int r = 0; r < 8; ++r) {
            float v = fmaxf(fmaxf(wmax[0][rbase + r], wmax[1][rbase + r]),
                            fmaxf(wmax[2][rbase + r], wmax[3][rbase + r]));
            dm[r] = v;
        }
        if (tid < 16) {
            Ofs[tid] += fmaxf(fmaxf(wmax[0][tid], wmax[1][tid]),
                              fmaxf(wmax[2][tid], wmax[3][tid]));
        }
        // exp args <= 0; hardware flush below 2^-126 is the desired behavior
        #pragma unroll
        for (int r = 0; r < 8; ++r)
            lds_w[(rbase + r) * WSTR + jc] =
                __builtin_amdgcn_exp2f((d[r] - dm[r]) * INV_LN2_F);
        #pragma unroll
        for (int r = 0; r < 8; ++r) xc[r] = xn[r];
        __syncthreads();   // w' visible for next step
    }

    // free energy + NLL contribution
    if (tid < 16) {
        float s = 0.0f;
        for (int j = 0; j < T_; ++j) s += lds_w[tid * WSTR + j];
        red[tid] = (Ofs[tid] + __logf(s)) - path_e[bblk + tid];
    }
    __syncthreads();
    if (tid == 0) {
        float s = 0.0f;
        for (int i = 0; i < 16; ++i) s += red[i];
        atomicAdd(out, s * (1.0f / (float)B_));
    }
}

// ---------------------------------------------------------------- launch ----
extern "C" void kernel_launch(void* const* d_in, const int* in_sizes, int n_in,
                              void* d_out, int out_size, void* d_ws, size_t ws_size,
                              hipStream_t stream) {
    const float* emis = (const float*)d_in[0];
    const int*   tags = (const int*)d_in[1];
    const float* U    = (const float*)d_in[2];
    const float* bs   = (const float*)d_in[3];
    const float* be   = (const float*)d_in[4];
    float* out = (float*)d_out;

    float* expU   = (float*)d_ws;          // 64*64 floats
    float* path_e = expU + T_ * T_;        // 256 floats

    crf_setup<<<1, 256, 0, stream>>>(U, expU, out);
    crf_path<<<B_, 256, 0, stream>>>(emis, tags, U, bs, be, path_e);
    crf_forward<<<B_ / 16, 128, 0, stream>>>(emis, expU, bs, be, path_e, out);
}